// NestedGIN_55946243998145
// MI455X (gfx1250) — compile-verified
//
#include <hip/hip_runtime.h>
#include <hip/hip_bf16.h>

// ---------------------------------------------------------------------------
// NestedGIN for MI455X (gfx1250, wave32).
// Dense layers use V_WMMA_F32_16X16X4_F32 (full-precision f32 WMMA), edge
// aggregation / pooling use global_atomic_add_f32. Memory-bound workload ->
// keep f32 end-to-end; WMMA f32 16x16x4 gives the matrix path at no precision
// loss.
// ---------------------------------------------------------------------------

typedef float v2f __attribute__((ext_vector_type(2)));
typedef float v8f __attribute__((ext_vector_type(8)));

#define HDIM 64

// ---------------------------------------------------------------------------
// GEMM: Out[M,64] = act( (A (+A2)) [M,64] @ W[64,64] + bias[64] )
// One block = 128 threads = 4 waves. Block handles 16 rows; wave w handles
// column tile n0 = 16*w. A tile staged in LDS (shared by the 4 waves).
// WMMA f32 16x16x4: A fragment v2f (lane<16: K=2h+0/1 of row lid),
// C/D: VGPR r -> row r (lanes 0-15) / row r+8 (lanes 16-31).
// ---------------------------------------------------------------------------
__global__ __launch_bounds__(128)
void gemm64_wmma(const float* __restrict__ A, const float* __restrict__ A2,
                 const float* __restrict__ W, const float* __restrict__ bias,
                 float* __restrict__ Out, int M, int do_relu)
{
    __shared__ float As[16 * HDIM];

    const int m0  = blockIdx.x * 16;
    const int tid = threadIdx.x;

    // Cooperative A-tile load (with optional fused elementwise add of A2).
    for (int i = tid; i < 16 * HDIM; i += 128) {
        const int r = i >> 6;
        const int c = i & 63;
        const int row = m0 + r;
        float v = 0.0f;
        if (row < M) {
            const long long gi = (long long)row * HDIM + c;
            v = A[gi];
            if (A2) v += A2[gi];
        }
        As[i] = v;
    }
    __syncthreads();

    const int wave = tid >> 5;       // 0..3 -> column tile
    const int lane = tid & 31;
    const int half = lane >> 4;      // 0 or 1
    const int lid  = lane & 15;
    const int n0   = wave * 16;

    v8f acc = {};
#pragma unroll
    for (int k0 = 0; k0 < HDIM; k0 += 4) {
        v2f a, b;
        const int ka = k0 + 2 * half;
        a.x = As[lid * HDIM + ka + 0];
        a.y = As[lid * HDIM + ka + 1];
        b.x = W[(ka + 0) * HDIM + n0 + lid];
        b.y = W[(ka + 1) * HDIM + n0 + lid];
        // D = A*B + C   (emits v_wmma_f32_16x16x4_f32)
        acc = __builtin_amdgcn_wmma_f32_16x16x4_f32(
            /*neg_a=*/false, a, /*neg_b=*/false, b,
            /*c_mod=*/(short)0, acc, /*reuse_a=*/false, /*reuse_b=*/false);
    }

    const int   col  = n0 + lid;
    const float bcol = bias[col];
#pragma unroll
    for (int r = 0; r < 8; ++r) {
        const int row = m0 + r + 8 * half;
        if (row < M) {
            float v = acc[r] + bcol;
            if (do_relu) v = fmaxf(v, 0.0f);
            Out[(long long)row * HDIM + col] = v;
        }
    }
}

// ---------------------------------------------------------------------------
// conv1 edge aggregation: x has feature dim 1.  agg1[dst] += x[src].
// ---------------------------------------------------------------------------
__global__ void edge_agg1(const int* __restrict__ src, const int* __restrict__ dst,
                          const float* __restrict__ x, float* __restrict__ agg1, int E)
{
    const int e = blockIdx.x * blockDim.x + threadIdx.x;
    if (e >= E) return;
    atomicAdd(&agg1[dst[e]], x[src[e]]);
}

// ---------------------------------------------------------------------------
// conv1 lift: t[i][j] = relu((x[i] + agg1[i]) * w1[j] + b1[j])  (1 -> 64)
// ---------------------------------------------------------------------------
__global__ void conv1_lift(const float* __restrict__ x, const float* __restrict__ agg1,
                           const float* __restrict__ w1, const float* __restrict__ b1,
                           float* __restrict__ t, int Nn)
{
    const int i = blockIdx.x * blockDim.x + threadIdx.x;
    const int node = i >> 6;
    if (node >= Nn) return;
    const int j = i & 63;
    const float v = x[node] + agg1[node];
    const float y = fmaf(v, w1[j], b1[j]);
    t[i] = fmaxf(y, 0.0f);
}

// ---------------------------------------------------------------------------
// H=64 edge aggregation: agg[dst][:] += h[src][:].  Thread per (edge, 4 feats).
// ---------------------------------------------------------------------------
__global__ void edge_aggH(const int* __restrict__ src, const int* __restrict__ dst,
                          const float* __restrict__ h, float* __restrict__ agg,
                          long long total)
{
    const long long i = (long long)blockIdx.x * blockDim.x + threadIdx.x;
    if (i >= total) return;
    const int e = (int)(i >> 4);
    const int f = (int)(i & 15) << 2;
    const int s = src[e];
    const int d = dst[e];
    const float4 v = *(const float4*)(h + (long long)s * HDIM + f);
    float* o = agg + (long long)d * HDIM + f;
    atomicAdd(o + 0, v.x);
    atomicAdd(o + 1, v.y);
    atomicAdd(o + 2, v.z);
    atomicAdd(o + 3, v.w);
}

// ---------------------------------------------------------------------------
// Segment-sum pooling: out[seg[r]][:] += h[r][:].  Thread per (row, 4 feats).
// ---------------------------------------------------------------------------
__global__ void pool_scatter(const int* __restrict__ seg, const float* __restrict__ h,
                             float* __restrict__ out, int rows)
{
    const long long i = (long long)blockIdx.x * blockDim.x + threadIdx.x;
    if (i >= (long long)rows * 16) return;
    const int r = (int)(i >> 4);
    const int f = (int)(i & 15) << 2;
    const int s = seg[r];
    const float4 v = *(const float4*)(h + (long long)r * HDIM + f);
    float* o = out + (long long)s * HDIM + f;
    atomicAdd(o + 0, v.x);
    atomicAdd(o + 1, v.y);
    atomicAdd(o + 2, v.z);
    atomicAdd(o + 3, v.w);
}

// ---------------------------------------------------------------------------
// Head: z = hg @ lin2_w + lin2_b ; out = log_softmax(z).  One thread / graph.
// ---------------------------------------------------------------------------
__global__ void head_kernel(const float* __restrict__ hg, const float* __restrict__ w2,
                            const float* __restrict__ b2, float* __restrict__ out,
                            int Gn, int OutDim)
{
    const int g = blockIdx.x * blockDim.x + threadIdx.x;
    if (g >= Gn) return;
    float z[8];
    for (int o = 0; o < OutDim; ++o) {
        float acc = b2[o];
        for (int k = 0; k < HDIM; ++k)
            acc = fmaf(hg[g * HDIM + k], w2[k * OutDim + o], acc);
        z[o] = acc;
    }
    float m = z[0];
    for (int o = 1; o < OutDim; ++o) m = fmaxf(m, z[o]);
    float s = 0.0f;
    for (int o = 0; o < OutDim; ++o) s += expf(z[o] - m);
    const float lse = logf(s);
    for (int o = 0; o < OutDim; ++o) out[g * OutDim + o] = z[o] - m - lse;
}

// ---------------------------------------------------------------------------
// Host-side orchestration (graph-capture safe: only kernel launches and
// hipMemsetAsync on `stream`).
// ---------------------------------------------------------------------------
extern "C" void kernel_launch(void* const* d_in, const int* in_sizes, int n_in,
                              void* d_out, int out_size, void* d_ws, size_t ws_size,
                              hipStream_t stream)
{
    // Inputs per setup_inputs() order.
    const float* x        = (const float*)d_in[0];   // (N,1)
    const int*   eidx     = (const int*)  d_in[1];   // (2,E)
    const int*   n2s      = (const int*)  d_in[2];   // (N,)
    const int*   s2g      = (const int*)  d_in[3];   // (S,)
    const float* c1_w1    = (const float*)d_in[4];   // (1,H)
    const float* c1_b1    = (const float*)d_in[5];   // (H,)
    const float* c1_w2    = (const float*)d_in[6];   // (H,H)
    const float* c1_b2    = (const float*)d_in[7];   // (H,)
    const float* cs_w1    = (const float*)d_in[8];   // (L,H,H)
    const float* cs_b1    = (const float*)d_in[9];   // (L,H)
    const float* cs_w2    = (const float*)d_in[10];  // (L,H,H)
    const float* cs_b2    = (const float*)d_in[11];  // (L,H)
    const float* lin1_w   = (const float*)d_in[12];  // (H,H)
    const float* lin1_b   = (const float*)d_in[13];  // (H,)
    const float* lin2_w   = (const float*)d_in[14];  // (H,OUT)
    const float* lin2_b   = (const float*)d_in[15];  // (OUT,)
    float*       out      = (float*)d_out;

    const int N   = in_sizes[0];          // 100000
    const int E   = in_sizes[1] / 2;      // 1600000
    const int S   = in_sizes[3];          // 2000
    const int OUT = in_sizes[15];         // 8
    const int G   = out_size / OUT;       // 64
    const int L   = in_sizes[9] / HDIM;   // 3

    const int* src = eidx;
    const int* dst = eidx + E;

    // Workspace carve-up (all f32).
    float* h0     = (float*)d_ws;                     // N*64  activations
    float* t      = h0     + (size_t)N * HDIM;        // N*64  mid-MLP
    float* agg    = t      + (size_t)N * HDIM;        // N*64  neighbor sums
    float* agg1   = agg    + (size_t)N * HDIM;        // N     conv1 dim-1 sums
    float* pool_s = agg1   + (size_t)N;               // S*64
    float* pool_g = pool_s + (size_t)S * HDIM;        // G*64
    float* hg1    = pool_g + (size_t)G * HDIM;        // G*64
    (void)ws_size; (void)n_in;

    const int gemmBlocksN = (N + 15) / 16;
    const int gemmBlocksG = (G + 15) / 16;
    const long long eTot  = (long long)E * 16;
    const int eBlocks     = (int)((eTot + 255) / 256);

    // ----- conv1: agg1 = segment_sum(x[src], dst) ; t = relu((x+agg1)w1+b1)
    hipMemsetAsync(agg1, 0, (size_t)N * sizeof(float), stream);
    edge_agg1<<<(E + 255) / 256, 256, 0, stream>>>(src, dst, x, agg1, E);
    conv1_lift<<<((size_t)N * HDIM + 255) / 256, 256, 0, stream>>>(x, agg1, c1_w1, c1_b1, t, N);
    // h0 = relu(t @ c1_w2 + c1_b2)
    gemm64_wmma<<<gemmBlocksN, 128, 0, stream>>>(t, nullptr, c1_w2, c1_b2, h0, N, 1);

    // ----- L more GIN layers
    for (int i = 0; i < L; ++i) {
        hipMemsetAsync(agg, 0, (size_t)N * HDIM * sizeof(float), stream);
        edge_aggH<<<eBlocks, 256, 0, stream>>>(src, dst, h0, agg, eTot);
        // t  = relu((h0 + agg) @ w1_i + b1_i)
        gemm64_wmma<<<gemmBlocksN, 128, 0, stream>>>(
            h0, agg, cs_w1 + (size_t)i * HDIM * HDIM, cs_b1 + (size_t)i * HDIM, t, N, 1);
        // h0 = relu(t @ w2_i + b2_i)
        gemm64_wmma<<<gemmBlocksN, 128, 0, stream>>>(
            t, nullptr, cs_w2 + (size_t)i * HDIM * HDIM, cs_b2 + (size_t)i * HDIM, h0, N, 1);
    }

    // ----- node -> subgraph -> graph double add-pool
    hipMemsetAsync(pool_s, 0, (size_t)S * HDIM * sizeof(float), stream);
    hipMemsetAsync(pool_g, 0, (size_t)G * HDIM * sizeof(float), stream);
    pool_scatter<<<(int)(((long long)N * 16 + 255) / 256), 256, 0, stream>>>(n2s, h0, pool_s, N);
    pool_scatter<<<(int)(((long long)S * 16 + 255) / 256), 256, 0, stream>>>(s2g, pool_s, pool_g, S);

    // ----- graph head: hg1 = relu(pool_g @ lin1_w + lin1_b)
    gemm64_wmma<<<gemmBlocksG, 128, 0, stream>>>(pool_g, nullptr, lin1_w, lin1_b, hg1, G, 1);
    // out = log_softmax(hg1 @ lin2_w + lin2_b)
    head_kernel<<<(G + 63) / 64, 64, 0, stream>>>(hg1, lin2_w, lin2_b, out, G, OUT);
}